// ContrastiveLoss_35399120454267
// MI455X (gfx1250) — compile-verified
//
#include <hip/hip_runtime.h>
#include <hip/hip_bf16.h>
#include <math.h>

// ---------------------------------------------------------------------------
// Contrastive loss, fused for MI455X (gfx1250, wave32, WMMA f32 16x16x4).
//   Kernel 1: per-row norms, pos-pair dot, JS-divergence row terms.
//   Kernel 2: fused GEMM (x@x^T and x@y^T) on v_wmma_f32_16x16x4_f32 with
//             async global->LDS double-buffered tiles and an
//             exp / diagonal-mask / row-sum epilogue -> per-coltile partials.
//   Kernel 3: deterministic reduce + cumulative "neg" + final scalar loss.
// ---------------------------------------------------------------------------

typedef float v2f __attribute__((ext_vector_type(2)));
typedef float v8f __attribute__((ext_vector_type(8)));
typedef int   v4i __attribute__((ext_vector_type(4)));

#define N_ROWS 4096
#define K_DIM  1024
#define T_INV_BASE 0.15f
#define EPS_C  1e-8f

// ---- CDNA5 async global->LDS copy (128-bit), with safe fallback -----------
#if defined(__has_builtin)
#if __has_builtin(__builtin_amdgcn_global_load_async_to_lds_b128)
#define HAVE_ASYNC_LDS 1
#endif
#endif

typedef __attribute__((address_space(1))) v4i* gas_v4i_p;
typedef __attribute__((address_space(3))) v4i* las_v4i_p;

__device__ __forceinline__ void async_cp16(float* lds, const float* g) {
#ifdef HAVE_ASYNC_LDS
    __builtin_amdgcn_global_load_async_to_lds_b128(
        (gas_v4i_p)g, (las_v4i_p)lds, 0, 0);
#else
    *reinterpret_cast<float4*>(lds) = *reinterpret_cast<const float4*>(g);
#endif
}

template <int N>
__device__ __forceinline__ void wait_async() {
#ifdef HAVE_ASYNC_LDS
#if __has_builtin(__builtin_amdgcn_s_wait_asynccnt)
    __builtin_amdgcn_s_wait_asynccnt(N);
#else
    asm volatile("s_wait_asynccnt %0" ::"i"(N));
#endif
#endif
}

// ---------------- Kernel 1: per-row statistics (1 wave per row) ------------
__global__ void cl_rows_kernel(const float* __restrict__ x,
                               const float* __restrict__ y,
                               float* __restrict__ nxg,
                               float* __restrict__ nyg,
                               float* __restrict__ dpos,
                               float* __restrict__ jsrow) {
    const int lane = threadIdx.x & 31;
    const int wave = threadIdx.x >> 5;
    const int row  = blockIdx.x * 8 + wave;

    const float* xr = x + (size_t)row * K_DIM;
    const float* yr = y + (size_t)row * K_DIM;

    float xv[32], yv[32];
#pragma unroll
    for (int t = 0; t < 32; ++t) {
        xv[t] = xr[t * 32 + lane];
        yv[t] = yr[t * 32 + lane];
    }

    float sx2 = 0.f, sy2 = 0.f, dxy = 0.f, mx = -1e30f, my = -1e30f;
#pragma unroll
    for (int t = 0; t < 32; ++t) {
        sx2 += xv[t] * xv[t];
        sy2 += yv[t] * yv[t];
        dxy += xv[t] * yv[t];
        mx = fmaxf(mx, xv[t]);
        my = fmaxf(my, yv[t]);
    }
#pragma unroll
    for (int m = 16; m >= 1; m >>= 1) {
        sx2 += __shfl_xor(sx2, m);
        sy2 += __shfl_xor(sy2, m);
        dxy += __shfl_xor(dxy, m);
        mx = fmaxf(mx, __shfl_xor(mx, m));
        my = fmaxf(my, __shfl_xor(my, m));
    }
    const float nx = sqrtf(sx2);
    const float ny = sqrtf(sy2);

    float sex = 0.f, sey = 0.f;
#pragma unroll
    for (int t = 0; t < 32; ++t) {
        sex += __expf(xv[t] - mx);
        sey += __expf(yv[t] - my);
    }
#pragma unroll
    for (int m = 16; m >= 1; m >>= 1) {
        sex += __shfl_xor(sex, m);
        sey += __shfl_xor(sey, m);
    }
    const float lsex = __logf(sex);
    const float lsey = __logf(sey);

    float kla = 0.f, klb = 0.f;
#pragma unroll
    for (int t = 0; t < 32; ++t) {
        const float la = (xv[t] - mx) - lsex;
        const float lb = (yv[t] - my) - lsey;
        const float a  = __expf(la);
        const float b  = __expf(lb);
        const float lm = __logf(0.5f * (a + b));
        kla += a * (la - lm);
        klb += b * (lb - lm);
    }
#pragma unroll
    for (int m = 16; m >= 1; m >>= 1) {
        kla += __shfl_xor(kla, m);
        klb += __shfl_xor(klb, m);
    }

    if (lane == 0) {
        nxg[row]   = nx;
        nyg[row]   = ny;
        dpos[row]  = dxy / fmaxf(nx * ny, EPS_C) / T_INV_BASE; // log(pos)
        jsrow[row] = kla + klb;
    }
}

// ---------------- Kernel 2: fused WMMA GEMM + exp epilogue ------------------
// Tile: 64 rows (i, from x) x 128 cols (j, from x AND y). 8 waves (2x4).
// Each wave: 32x32 region = 2x2 WMMA 16x16 tiles for Sxx and for Sxy.
// K tiled by 32 through double-buffered LDS filled by async copies.
#define BM 64
#define BN 128
#define BK 32
#define NCHUNK (K_DIM / BK)
#define LDSW 36   // padded row stride (floats), 144B: 16B-aligned, conflict-free

// 10 async b128 copies per thread per chunk (A:2, Bx:4, By:4)
__device__ __forceinline__ void issue_tile_loads(const float* __restrict__ x,
                                                 const float* __restrict__ y,
                                                 float* As, float* Bx, float* By,
                                                 int tid, int rowBase, int colBase,
                                                 int k0) {
#pragma unroll
    for (int q = 0; q < 2; ++q) {
        const int idx = tid + 256 * q;
        const int row = idx >> 3;
        const int c4  = (idx & 7) * 4;
        async_cp16(&As[row * LDSW + c4],
                   x + (size_t)(rowBase + row) * K_DIM + k0 + c4);
    }
#pragma unroll
    for (int q = 0; q < 4; ++q) {
        const int idx = tid + 256 * q;
        const int row = idx >> 3;
        const int c4  = (idx & 7) * 4;
        const size_t g = (size_t)(colBase + row) * K_DIM + k0 + c4;
        async_cp16(&Bx[row * LDSW + c4], x + g);
        async_cp16(&By[row * LDSW + c4], y + g);
    }
}

__global__ void __launch_bounds__(256)
cl_gemm_kernel(const float* __restrict__ x,
               const float* __restrict__ y,
               const float* __restrict__ nxg,
               const float* __restrict__ nyg,
               float* __restrict__ partials) {
    __shared__ float As[2][BM * LDSW];
    __shared__ float Bx[2][BN * LDSW];
    __shared__ float By[2][BN * LDSW];
    __shared__ float lnxi[BM];
    __shared__ float lnxj[BN];
    __shared__ float lnyj[BN];
    __shared__ float rowpart[4][BM];

    const int tid  = threadIdx.x;
    const int lane = tid & 31;
    const int wave = tid >> 5;
    const int wr   = wave >> 2;   // 0..1  (row group of 32)
    const int wc   = wave & 3;    // 0..3  (col group of 32)
    const int r0   = wr * 32;
    const int c0   = wc * 32;
    const int lr   = lane & 15;
    const int half = lane >> 4;

    const int colBase = blockIdx.x * BN;  // 32 col tiles
    const int rowBase = blockIdx.y * BM;  // 64 row tiles

    // preload norms for this tile (covered by first barrier below)
    if (tid < BM) lnxi[tid] = nxg[rowBase + tid];
    if (tid < BN) { lnxj[tid] = nxg[colBase + tid]; lnyj[tid] = nyg[colBase + tid]; }

    v8f axx[2][2] = {};
    v8f axy[2][2] = {};

    // prologue: fill buffer 0
    issue_tile_loads(x, y, As[0], Bx[0], By[0], tid, rowBase, colBase, 0);

    for (int c = 0; c < NCHUNK; ++c) {
        const int cur = c & 1;
        if (c + 1 < NCHUNK) {
            // overlap: start next chunk's copies, then wait only for chunk c
            issue_tile_loads(x, y, As[cur ^ 1], Bx[cur ^ 1], By[cur ^ 1],
                             tid, rowBase, colBase, (c + 1) * BK);
            wait_async<10>();   // in-order: <=10 outstanding => chunk c landed
        } else {
            wait_async<0>();
        }
        __syncthreads();

        const float* Asc = As[cur];
        const float* Bxc = Bx[cur];
        const float* Byc = By[cur];
#pragma unroll
        for (int kk = 0; kk < BK; kk += 4) {
            const int koff = kk + 2 * half;  // f32 frag: VGPR0=K{0,2}, VGPR1=K{1,3}
            v2f a[2], bx[2], by[2];
#pragma unroll
            for (int mt = 0; mt < 2; ++mt)
                a[mt] = *reinterpret_cast<const v2f*>(
                    &Asc[(r0 + 16 * mt + lr) * LDSW + koff]);
#pragma unroll
            for (int nt = 0; nt < 2; ++nt) {
                bx[nt] = *reinterpret_cast<const v2f*>(
                    &Bxc[(c0 + 16 * nt + lr) * LDSW + koff]);
                by[nt] = *reinterpret_cast<const v2f*>(
                    &Byc[(c0 + 16 * nt + lr) * LDSW + koff]);
            }
#pragma unroll
            for (int mt = 0; mt < 2; ++mt)
#pragma unroll
                for (int nt = 0; nt < 2; ++nt) {
                    axx[mt][nt] = __builtin_amdgcn_wmma_f32_16x16x4_f32(
                        false, a[mt], false, bx[nt], (short)0, axx[mt][nt], false, false);
                    axy[mt][nt] = __builtin_amdgcn_wmma_f32_16x16x4_f32(
                        false, a[mt], false, by[nt], (short)0, axy[mt][nt], false, false);
                }
        }
        __syncthreads();  // all waves done reading buf[cur] before its refill
    }

    // Epilogue: exp(S / (max(n_i*n_j,eps)*T)), mask i==j, row-reduce.
    // C/D layout: VGPR r, lanes0-15 -> (M=r, N=lane); lanes16-31 -> (M=8+r, N=lane-16)
#pragma unroll
    for (int mt = 0; mt < 2; ++mt) {
#pragma unroll
        for (int r = 0; r < 8; ++r) {
            const int li  = r0 + 16 * mt + r + 8 * half;  // local row 0..63
            const int gi  = rowBase + li;
            const float ni = lnxi[li];
            float e = 0.f;
#pragma unroll
            for (int nt = 0; nt < 2; ++nt) {
                const int jl = c0 + 16 * nt + lr;
                const int gj = colBase + jl;
                if (gi != gj) {
                    const float sxx = axx[mt][nt][r];
                    const float sxy = axy[mt][nt][r];
                    e += __expf(sxx / (fmaxf(ni * lnxj[jl], EPS_C) * T_INV_BASE));
                    e += __expf(sxy / (fmaxf(ni * lnyj[jl], EPS_C) * T_INV_BASE));
                }
            }
#pragma unroll
            for (int m = 1; m <= 8; m <<= 1) e += __shfl_xor(e, m);
            if (lr == 0) rowpart[wc][li] = e;   // unique slot -> deterministic
        }
    }
    __syncthreads();

    if (tid < BM) {
        float p = rowpart[0][tid] + rowpart[1][tid] + rowpart[2][tid] + rowpart[3][tid];
        partials[(size_t)blockIdx.x * N_ROWS + rowBase + tid] = p;
    }
}

// ---------------- Kernel 3: reduce + cumulative neg + loss ------------------
__global__ void cl_finalize_kernel(const float* __restrict__ dpos,
                                   const float* __restrict__ jsrow,
                                   const float* __restrict__ partials,
                                   float* __restrict__ out) {
    __shared__ float rs[N_ROWS];
    __shared__ float sd[256];
    __shared__ float sj[256];
    const int tid = threadIdx.x;

    float locd = 0.f, locj = 0.f;
    for (int row = tid; row < N_ROWS; row += 256) {
        float s = 0.f;
        for (int c = 0; c < 32; ++c) s += partials[(size_t)c * N_ROWS + row];
        rs[row] = s;
        locd += dpos[row];
        locj += jsrow[row];
    }
    sd[tid] = locd;
    sj[tid] = locj;
    __syncthreads();
    for (int s = 128; s >= 1; s >>= 1) {
        if (tid < s) { sd[tid] += sd[tid + s]; sj[tid] += sj[tid + s]; }
        __syncthreads();
    }
    if (tid == 0) {
        // faithful to source bug: accumulator never reset -> cumulative neg
        float run = 0.f, slog = 0.f;
        for (int i = 0; i < N_ROWS; ++i) {
            run += rs[i];
            slog += __logf(run);
        }
        const float nce = slog - sd[0];                 // -sum(log(pos/neg))
        const float js  = 0.5f * sj[0] / (float)N_ROWS; // batchmean JS
        out[0] = nce + 1.0f * js;
    }
}

// ---------------------------------------------------------------------------
extern "C" void kernel_launch(void* const* d_in, const int* in_sizes, int n_in,
                              void* d_out, int out_size, void* d_ws, size_t ws_size,
                              hipStream_t stream) {
    (void)in_sizes; (void)n_in; (void)out_size; (void)ws_size;
    const float* x = (const float*)d_in[0];
    const float* y = (const float*)d_in[1];
    float* out = (float*)d_out;

    float* ws       = (float*)d_ws;
    float* nxg      = ws;                       // 4096
    float* nyg      = ws + N_ROWS;              // 4096
    float* dpos     = ws + 2 * N_ROWS;          // 4096
    float* jsrow    = ws + 3 * N_ROWS;          // 4096
    float* partials = ws + 4 * N_ROWS;          // 32 * 4096

    cl_rows_kernel<<<N_ROWS / 8, 256, 0, stream>>>(x, y, nxg, nyg, dpos, jsrow);
    cl_gemm_kernel<<<dim3(N_ROWS / BN, N_ROWS / BM), 256, 0, stream>>>(
        x, y, nxg, nyg, partials);
    cl_finalize_kernel<<<1, 256, 0, stream>>>(dpos, jsrow, partials, out);
}